// ScalingModel_60284160967022
// MI455X (gfx1250) — compile-verified
//
#include <hip/hip_runtime.h>
#include <hip/hip_bf16.h>

typedef __attribute__((ext_vector_type(2))) float v2f;
typedef __attribute__((ext_vector_type(8))) float v8f;

#define WMMA4(a, bfrag, c) \
  __builtin_amdgcn_wmma_f32_16x16x4_f32(false, (a), false, (bfrag), (short)0, (c), false, false)

namespace {
constexpr int BB = 16, JJ = 256, KK = 384, CMETA = 6, WID = 32, SAMP = 32, NDEPTH = 3;
constexpr int PB3 = JJ * KK;        // 98304 points per batch (scale branch)
constexpr int PB2 = SAMP * KK;      // 12288 points per batch (image branch)
constexpr int HSTR = 36;            // padded LDS row stride (floats)
constexpr int IMG_CHUNK = 2048;     // points per WG, image branch  -> 6 WGs / batch
constexpr int SCL_CHUNK = 4096;     // points per WG, scale branch  -> 24 WGs / batch
constexpr int WVB = (PB2 / IMG_CHUNK) * 8;  // 48 wave-partials per batch
}

// ---------------------------------------------------------------------------
// JAX threefry2x32 (key = PRNGKey(42) -> (0,42)), gumbel for (16,256) shape
// ---------------------------------------------------------------------------
__device__ __forceinline__ unsigned rotl32(unsigned x, unsigned n) {
  return (x << n) | (x >> (32u - n));
}

__device__ void threefry2x32_42(unsigned x0, unsigned x1, unsigned& o0, unsigned& o1) {
  const unsigned ks[3] = {0u, 42u, 0u ^ 42u ^ 0x1BD11BDAu};
  const unsigned rotA[4] = {13u, 15u, 26u, 6u};
  const unsigned rotB[4] = {17u, 29u, 16u, 24u};
  x0 += ks[0]; x1 += ks[1];
#pragma unroll
  for (int i = 0; i < 5; ++i) {
    const unsigned* r = (i & 1) ? rotB : rotA;
#pragma unroll
    for (int jr = 0; jr < 4; ++jr) { x0 += x1; x1 = rotl32(x1, r[jr]); x1 ^= x0; }
    x0 += ks[(i + 1) % 3];
    x1 += ks[(i + 2) % 3] + (unsigned)(i + 1);
  }
  o0 = x0; o1 = x1;
}

__device__ float gumbel_at(int m) {  // m in [0, 4096)
  unsigned i = (unsigned)(m & 2047);
  unsigned r0, r1;
  threefry2x32_42(i, i + 2048u, r0, r1);
  unsigned bits = (m < 2048) ? r0 : r1;
  float f = __uint_as_float((bits >> 9) | 0x3f800000u) - 1.0f;
  const float tiny = 1.1754944e-38f;
  float u = fmaxf(tiny, f * (1.0f - tiny) + tiny);
  return -__logf(-__logf(u));
}

// ---------------------------------------------------------------------------
// Phase 1: p = any(mask,K); scores = log(p/sum)+gumbel; top-32 j per batch
// ---------------------------------------------------------------------------
__global__ void __launch_bounds__(256) k_sample(const unsigned char* __restrict__ mask,
                                                int* __restrict__ idxOut) {
  __shared__ float sc[JJ];
  __shared__ float red[JJ];
  __shared__ int redi[JJ];
  const int b = blockIdx.x, j = threadIdx.x;

  const uint4* mp = (const uint4*)(mask + (size_t)(b * JJ + j) * KK);
  unsigned any = 0;
#pragma unroll
  for (int t = 0; t < KK / 16; ++t) { uint4 v = mp[t]; any |= v.x | v.y | v.z | v.w; }
  float pv = any ? 1.0f : 0.0f;

  red[j] = pv; __syncthreads();
  for (int s = JJ / 2; s > 0; s >>= 1) { if (j < s) red[j] += red[j + s]; __syncthreads(); }
  float psum = red[0]; __syncthreads();

  sc[j] = (pv > 0.0f) ? (-__logf(psum) + gumbel_at(b * JJ + j)) : -__builtin_inff();
  __syncthreads();

  for (int pick = 0; pick < SAMP; ++pick) {
    red[j] = sc[j]; redi[j] = j; __syncthreads();
    for (int s = JJ / 2; s > 0; s >>= 1) {
      if (j < s) {
        if (red[j + s] > red[j] || (red[j + s] == red[j] && redi[j + s] < redi[j])) {
          red[j] = red[j + s]; redi[j] = redi[j + s];
        }
      }
      __syncthreads();
    }
    if (j == 0) { idxOut[b * SAMP + pick] = redi[0]; sc[redi[0]] = -__builtin_inff(); }
    __syncthreads();
  }
}

// ---------------------------------------------------------------------------
// B-fragment loader for V_WMMA_F32_16X16X4_F32 (B is KxN = 4x16):
// lanes 0-15 hold rows k0+0/k0+1, lanes 16-31 hold rows k0+2/k0+3
// ---------------------------------------------------------------------------
__device__ __forceinline__ v2f load_bfrag(const float* W, int k0, int col, int hl) {
  int kr = k0 + hl * 2;
  v2f r; r.x = W[kr * WID + col]; r.y = W[(kr + 1) * WID + col];
  return r;
}

// ---------------------------------------------------------------------------
// Phase 2: image branch -> per-wave masked partial sums of final activations
// ---------------------------------------------------------------------------
__global__ void __launch_bounds__(256, 1)
k_image(const float* __restrict__ Imodel, const float* __restrict__ Iobs,
        const float* __restrict__ metadata, const unsigned char* __restrict__ mask,
        const float* __restrict__ Wimg_in, const float* __restrict__ bimg_in,
        const float* __restrict__ Wimg, const float* __restrict__ bimg,
        const int* __restrict__ idxIn, float* __restrict__ wavePart) {
  __shared__ float Hall[8 * 16 * HSTR];
  const int b = blockIdx.y, chunk = blockIdx.x;
  const int tid = threadIdx.x, wv = tid >> 5, lane = tid & 31;
  const int m = lane & 15, hl = lane >> 4;
  float* H = Hall + wv * (16 * HSTR);
  const int col0 = m, col1 = m + 16;

  // invariant weight fragments (registers)
  v2f fin[2][2];
#pragma unroll
  for (int s = 0; s < 2; ++s) {
    fin[s][0] = load_bfrag(Wimg_in, 4 * s, col0, hl);
    fin[s][1] = load_bfrag(Wimg_in, 4 * s, col1, hl);
  }
  v2f fh[NDEPTH][8][2];
#pragma unroll
  for (int d = 0; d < NDEPTH; ++d)
#pragma unroll
    for (int s = 0; s < 8; ++s) {
      fh[d][s][0] = load_bfrag(Wimg + d * WID * WID, 4 * s, col0, hl);
      fh[d][s][1] = load_bfrag(Wimg + d * WID * WID, 4 * s, col1, hl);
    }
  const float bin0 = bimg_in[col0], bin1 = bimg_in[col1];
  float bh0[NDEPTH], bh1[NDEPTH];
#pragma unroll
  for (int d = 0; d < NDEPTH; ++d) { bh0[d] = bimg[d * WID + col0]; bh1[d] = bimg[d * WID + col1]; }

  float acc0 = 0.f, acc1 = 0.f, cacc = 0.f;

  for (int i = wv; i < IMG_CHUNK / 16; i += 8) {
    const int q0 = chunk * IMG_CHUNK + i * 16;       // tile never straddles a sampled row
    const int s_idx = q0 / KK, k0 = q0 % KK;
    const int jrow = idxIn[b * SAMP + s_idx];
    const size_t e0 = (size_t)((b * JJ + jrow) * KK + k0);
    const size_t e = e0 + (size_t)m;

    // A fragments: channels [im, io, md0..md5]
    const v2f* md2 = (const v2f*)(metadata + e * CMETA);
    v2f a0, a1;
    if (hl == 0) { a0.x = Imodel[e]; a0.y = Iobs[e]; a1 = md2[1]; }   // ch0,1 | ch4,5->md2,md3
    else         { a0 = md2[0];                      a1 = md2[2]; }   // ch2,3 | ch6,7->md4,md5

    v8f c0 = {0.f,0.f,0.f,0.f,0.f,0.f,0.f,0.f}, c1 = c0;
    c0 = WMMA4(a0, fin[0][0], c0); c0 = WMMA4(a1, fin[1][0], c0);
    c1 = WMMA4(a0, fin[0][1], c1); c1 = WMMA4(a1, fin[1][1], c1);
#pragma unroll
    for (int r = 0; r < 8; ++r) { c0[r] += bin0; c1[r] += bin1; }     // h0: affine only

#pragma unroll
    for (int d = 0; d < NDEPTH; ++d) {
#pragma unroll
      for (int r = 0; r < 8; ++r) {                                   // C-layout -> LDS
        H[(hl * 8 + r) * HSTR + col0] = c0[r];
        H[(hl * 8 + r) * HSTR + col1] = c1[r];
      }
      v8f n0 = {0.f,0.f,0.f,0.f,0.f,0.f,0.f,0.f}, n1 = n0;
#pragma unroll
      for (int s = 0; s < 8; ++s) {                                   // A from LDS (transposed)
        const float* ap = H + m * HSTR + 4 * s + hl * 2;
        v2f a; a.x = ap[0]; a.y = ap[1];
        n0 = WMMA4(a, fh[d][s][0], n0);
        n1 = WMMA4(a, fh[d][s][1], n1);
      }
#pragma unroll
      for (int r = 0; r < 8; ++r) {
        c0[r] = fmaxf(n0[r] + bh0[d], 0.f);
        c1[r] = fmaxf(n1[r] + bh1[d], 0.f);
      }
    }

    // masked accumulate: lane's 8 rows are [hl*8 .. hl*8+7]
    const unsigned long long mb = *(const unsigned long long*)(mask + e0 + hl * 8);
#pragma unroll
    for (int r = 0; r < 8; ++r) {
      float mf = ((mb >> (8 * r)) & 0xffull) ? 1.0f : 0.0f;
      acc0 += mf * c0[r]; acc1 += mf * c1[r]; cacc += mf;
    }
  }

  acc0 += __shfl_xor(acc0, 16);
  acc1 += __shfl_xor(acc1, 16);
#pragma unroll
  for (int off = 16; off > 0; off >>= 1) cacc += __shfl_xor(cacc, off);

  float* wp = wavePart + (size_t)(b * WVB + chunk * 8 + wv) * 33;
  if (hl == 0) { wp[col0] = acc0; wp[col1] = acc1; }
  if (lane == 0) wp[32] = cacc * (1.0f / 16.0f);  // each mask value counted 16x
}

// Phase 2.5: deterministic fixed-order reduction of the 48 wave partials
__global__ void k_reduce(const float* __restrict__ wavePart, float* __restrict__ rep) {
  const int b = blockIdx.x, t = threadIdx.x;
  if (t < 33) {
    float s = 0.f;
    for (int w = 0; w < WVB; ++w) s += wavePart[(size_t)(b * WVB + w) * 33 + t];
    rep[b * 33 + t] = s;
  }
}

// ---------------------------------------------------------------------------
// Phase 3: scale branch over all B*J*K points
// ---------------------------------------------------------------------------
__global__ void __launch_bounds__(256, 1)
k_scale(const float* __restrict__ metadata, const unsigned char* __restrict__ mask,
        const float* __restrict__ Wlin, const float* __restrict__ blin,
        const float* __restrict__ Wmlp, const float* __restrict__ bmlp,
        const float* __restrict__ Wout, const float* __restrict__ bout,
        const float* __restrict__ rep, float* __restrict__ out) {
  __shared__ float bias0s[WID];
  __shared__ float Hall[8 * 16 * HSTR];
  const int b = blockIdx.y, chunk = blockIdx.x;
  const int tid = threadIdx.x, wv = tid >> 5, lane = tid & 31;
  const int m = lane & 15, hl = lane >> 4;
  float* H = Hall + wv * (16 * HSTR);
  const int col0 = m, col1 = m + 16;

  if (tid < WID) {
    bias0s[tid] = blin[tid] + rep[b * 33 + tid] / rep[b * 33 + 32];  // + image_rep
  }

  // invariant weight fragments
  v2f fin[2][2];
  fin[0][0] = load_bfrag(Wlin, 0, col0, hl);
  fin[0][1] = load_bfrag(Wlin, 0, col1, hl);
  if (hl == 0) {  // rows 4,5 exist; rows 6,7 are zero padding
    fin[1][0] = load_bfrag(Wlin, 4, col0, 0);
    fin[1][1] = load_bfrag(Wlin, 4, col1, 0);
  } else {
    fin[1][0].x = 0.f; fin[1][0].y = 0.f; fin[1][1] = fin[1][0];
  }
  v2f fh[NDEPTH][8][2];
#pragma unroll
  for (int d = 0; d < NDEPTH; ++d)
#pragma unroll
    for (int s = 0; s < 8; ++s) {
      fh[d][s][0] = load_bfrag(Wmlp + d * WID * WID, 4 * s, col0, hl);
      fh[d][s][1] = load_bfrag(Wmlp + d * WID * WID, 4 * s, col1, hl);
    }
  float bh0[NDEPTH], bh1[NDEPTH];
#pragma unroll
  for (int d = 0; d < NDEPTH; ++d) { bh0[d] = bmlp[d * WID + col0]; bh1[d] = bmlp[d * WID + col1]; }
  const float bo = bout[0];
  // register-resident output weights (removes 32 ds loads per tile)
  float wr[WID];
#pragma unroll
  for (int n = 0; n < WID; ++n) wr[n] = Wout[n];
  __syncthreads();
  const float b0v0 = bias0s[col0], b0v1 = bias0s[col1];

  const size_t base = (size_t)b * PB3 + (size_t)chunk * SCL_CHUNK;

  for (int i = wv; i < SCL_CHUNK / 16; i += 8) {
    const size_t e0 = base + (size_t)i * 16;
    const size_t e = e0 + (size_t)m;
    const v2f* md2 = (const v2f*)(metadata + e * CMETA);
    __builtin_prefetch((const void*)(metadata + (e + 128) * CMETA), 0, 0);  // next tile for this wave

    v2f a0, a1;
    if (hl == 0) { a0 = md2[0]; a1 = md2[2]; }           // ch0,1 | ch4,5
    else         { a0 = md2[1]; a1.x = 0.f; a1.y = 0.f; }// ch2,3 | ch6,7=0

    v8f c0 = {0.f,0.f,0.f,0.f,0.f,0.f,0.f,0.f}, c1 = c0;
    c0 = WMMA4(a0, fin[0][0], c0); c0 = WMMA4(a1, fin[1][0], c0);
    c1 = WMMA4(a0, fin[0][1], c1); c1 = WMMA4(a1, fin[1][1], c1);
#pragma unroll
    for (int r = 0; r < 8; ++r) { c0[r] += b0v0; c1[r] += b0v1; }  // affine + image_rep

#pragma unroll
    for (int d = 0; d < NDEPTH; ++d) {
#pragma unroll
      for (int r = 0; r < 8; ++r) {
        H[(hl * 8 + r) * HSTR + col0] = c0[r];
        H[(hl * 8 + r) * HSTR + col1] = c1[r];
      }
      v8f n0 = {0.f,0.f,0.f,0.f,0.f,0.f,0.f,0.f}, n1 = n0;
#pragma unroll
      for (int s = 0; s < 8; ++s) {
        const float* ap = H + m * HSTR + 4 * s + hl * 2;
        v2f a; a.x = ap[0]; a.y = ap[1];
        n0 = WMMA4(a, fh[d][s][0], n0);
        n1 = WMMA4(a, fh[d][s][1], n1);
      }
#pragma unroll
      for (int r = 0; r < 8; ++r) {
        c0[r] = fmaxf(n0[r] + bh0[d], 0.f);
        c1[r] = fmaxf(n1[r] + bh1[d], 0.f);
      }
    }

    // final activations -> LDS, then 32->1 dot (lanes 0-15 own one point each)
#pragma unroll
    for (int r = 0; r < 8; ++r) {
      H[(hl * 8 + r) * HSTR + col0] = c0[r];
      H[(hl * 8 + r) * HSTR + col1] = c1[r];
    }
    float o = bo;
#pragma unroll
    for (int n = 0; n < WID; ++n) o += H[m * HSTR + n] * wr[n];
    if (hl == 0) {
      out[e0 + m] = mask[e0 + m] ? o : 0.0f;
    }
  }
}

// ---------------------------------------------------------------------------
extern "C" void kernel_launch(void* const* d_in, const int* in_sizes, int n_in,
                              void* d_out, int out_size, void* d_ws, size_t ws_size,
                              hipStream_t stream) {
  const float* Imodel = (const float*)d_in[0];
  const float* Iobs = (const float*)d_in[1];
  const float* metadata = (const float*)d_in[2];
  const unsigned char* mask = (const unsigned char*)d_in[3];  // jnp bool = 1 byte
  // d_in[4] = sample_size (== 32, baked in as SAMP)
  const float* Wimg_in = (const float*)d_in[5];
  const float* bimg_in = (const float*)d_in[6];
  const float* Wimg = (const float*)d_in[7];
  const float* bimg = (const float*)d_in[8];
  const float* Wlin = (const float*)d_in[9];
  const float* blin = (const float*)d_in[10];
  const float* Wmlp = (const float*)d_in[11];
  const float* bmlp = (const float*)d_in[12];
  const float* Wout = (const float*)d_in[13];
  const float* bout = (const float*)d_in[14];
  float* out = (float*)d_out;

  // workspace layout
  int* idxWs = (int*)d_ws;                                        // 16*32 ints
  float* wavePart = (float*)((char*)d_ws + 4096);                 // 16*48*33 floats
  float* rep = (float*)((char*)d_ws + 4096 + (size_t)BB * WVB * 33 * 4);  // 16*33 floats

  k_sample<<<BB, JJ, 0, stream>>>(mask, idxWs);
  k_image<<<dim3(PB2 / IMG_CHUNK, BB), 256, 0, stream>>>(
      Imodel, Iobs, metadata, mask, Wimg_in, bimg_in, Wimg, bimg, idxWs, wavePart);
  k_reduce<<<BB, 64, 0, stream>>>(wavePart, rep);
  k_scale<<<dim3(PB3 / SCL_CHUNK, BB), 256, 0, stream>>>(
      metadata, mask, Wlin, blin, Wmlp, bmlp, Wout, bout, rep, out);
}